// Model_80023830659464
// MI455X (gfx1250) — compile-verified
//
#include <hip/hip_runtime.h>

typedef __attribute__((ext_vector_type(2))) float v2f;
typedef __attribute__((ext_vector_type(8))) float v8f;
typedef __attribute__((ext_vector_type(4))) unsigned int u32x4;
typedef __attribute__((ext_vector_type(8))) int i32x8;
typedef __attribute__((ext_vector_type(4))) int i32x4;

// Problem constants
#define B_ROWS 2048
#define K_DIM  4800
#define WIN    48
#define RES    50
#define PRED   96
#define NPAD   64

// Workspace layout (float offsets)
#define OFF_M     0u        // 50*48   = 2400
#define OFF_DPOW  2560u     // 50*101  = 5050
#define OFF_KFRAG 8192u     // 1200*4*64 = 307200
#define OFF_PPART 315392u   // 10*2048*64 = 1310720
#define OFF_P     1626112u  // 2048*64 = 131072
#define OFF_LAST  1757184u  // 2048*64 = 131072

// GEMM config
#define MGRP_ROWS 64            // 4 waves * 16 rows
#define NUM_MGRP  32            // 2048/64
#define KSPLIT    10
#define KRANGE    480           // 4800/10
#define KC        32            // K elements per LDS chunk
#define NCHUNK    15            // 480/32
#define LDS_STRIDE 36           // 32 dwords + 4 dwords TDM pad (bank-conflict-free)

// TDM group-1 flag words: data_size=4B (code 2)
// x-tile adds: pad_enable | pad_interval=32dw (code 4) | pad_amount=4dw (code 3)
#define G1_XTILE  ((2u << 16) | (1u << 20) | (4u << 22) | (3u << 25))
#define G1_LINEAR (2u << 16)

// ---------------------------------------------------------------------------
// Issue one TDM tensor load (2D tile, or 1D when tile_d1 == 0).
// D# group0: count | lds_addr | global_addr | type=2
// D# group1: flags | tensor_dim0/1 | tile_dim0/1 | dim0_stride
// Groups 2/3 (dims 2..4) unused -> zeros. 6-arg builtin (amdgpu-toolchain).
// ---------------------------------------------------------------------------
__device__ __forceinline__ void tdm_load(unsigned lds_off, const void* gptr,
                                         unsigned tile_d0, unsigned tile_d1,
                                         unsigned tensor_d0, unsigned tensor_d1,
                                         unsigned long long stride0,
                                         unsigned flags) {
    unsigned long long ga = (unsigned long long)(uintptr_t)gptr;
    u32x4 g0;
    g0.x = 1u;                                   // count=1, user descriptor
    g0.y = lds_off;                              // LDS byte address
    g0.z = (unsigned)ga;                         // global_addr[31:0]
    g0.w = (unsigned)(ga >> 32) | (2u << 30);    // global_addr[56:32] | type=2
    i32x8 g1;
    g1[0] = (int)flags;                                            // wg_mask=0 | data_size | pad cfg
    g1[1] = (int)((tensor_d0 & 0xFFFFu) << 16);                    // [63:48] = dim0[15:0]
    g1[2] = (int)((tensor_d0 >> 16) | ((tensor_d1 & 0xFFFFu) << 16));
    g1[3] = (int)((tensor_d1 >> 16) | (tile_d0 << 16));            // [127:112] = tile_dim0
    g1[4] = (int)tile_d1;                                          // tile_dim1, tile_dim2=0
    g1[5] = (int)(unsigned)stride0;                                // dim0_stride[31:0]
    g1[6] = (int)((unsigned)(stride0 >> 32) & 0xFFFFu);            // dim0_stride[47:32]
    g1[7] = 0;
    i32x4 z4 = {0, 0, 0, 0};
    i32x8 z8 = {0, 0, 0, 0, 0, 0, 0, 0};
    __builtin_amdgcn_tensor_load_to_lds(g0, g1, z4, z4, z8, 0);
}

// ---------------------------------------------------------------------------
// K0: fold M = W_in @ W_lin (50x48) and build d-power table d^e, e=0..100
// ---------------------------------------------------------------------------
__global__ void k_prep(const float* __restrict__ W_lin,
                       const float* __restrict__ W_in,
                       const float* __restrict__ d,
                       float* __restrict__ ws) {
    float* M    = ws + OFF_M;
    float* dpow = ws + OFF_DPOW;
    int t = threadIdx.x;
    for (int idx = t; idx < RES * WIN; idx += 256) {
        int r = idx / WIN, w = idx % WIN;
        float s = 0.f;
#pragma unroll 8
        for (int v = 0; v < WIN; ++v)
            s = fmaf(W_in[r * WIN + v], W_lin[v * WIN + w], s);
        M[idx] = s;
    }
    if (t < RES) {
        float base = d[t];
        float pw = 1.f;
        for (int e = 0; e <= 100; ++e) { dpow[t * 101 + e] = pw; pw *= base; }
    }
}

// ---------------------------------------------------------------------------
// K1: materialize K (padded to 64 rows) directly in WMMA B-fragment layout.
// Fragment flat index: ((kstep*4 + ntile)*32 + lane)*2 + e
//   holds B element (K = 4*kstep + 2*(lane>>4) + e, N = ntile*16 + (lane&15))
//   value = d[n]^(99-j) * M[n][w]  with  k = j*48 + w
// ---------------------------------------------------------------------------
__global__ void k_kfrag(float* __restrict__ ws) {
    const float* M    = ws + OFF_M;
    const float* dpow = ws + OFF_DPOW;
    float* kfrag      = ws + OFF_KFRAG;
    int g = (blockIdx.x * 256 + threadIdx.x) * 4;
#pragma unroll
    for (int fi = 0; fi < 4; ++fi) {
        int flat  = g + fi;
        int e     = flat & 1;
        int pair  = flat >> 1;
        int lane  = pair & 31;
        int fn    = pair >> 5;
        int nt    = fn & 3;
        int kstep = fn >> 2;
        int n     = nt * 16 + (lane & 15);
        int k     = kstep * 4 + ((lane >> 4) << 1) + e;
        int j     = k / WIN;
        int w     = k - j * WIN;
        float val = 0.f;
        if (n < RES) val = dpow[n * 101 + (99 - j)] * M[n * WIN + w];
        kfrag[flat] = val;
    }
}

// ---------------------------------------------------------------------------
// K2: main GEMM  pPart[kg] = X[64-row group] @ K^T  via v_wmma_f32_16x16x4_f32
// TDM (tensor_load_to_lds) stages both the x tile (with HW LDS padding -> row
// stride 36 dwords) and the K-fragment chunk; double-buffered, TENSORcnt sync.
// ---------------------------------------------------------------------------
__global__ __launch_bounds__(128)
void k_gemm(const float* __restrict__ x,
            const float* __restrict__ kfrag,
            float* __restrict__ pPart) {
    __shared__ __align__(16) float ldsA[2][MGRP_ROWS * LDS_STRIDE]; // 2*64*36*4B = 18 KB
    __shared__ __align__(16) float ldsB[2][(KC / 4) * 4 * NPAD];    // 2*2048*4B  = 16 KB

    const int tid   = threadIdx.x;
    const int lane  = tid & 31;
    const int wave  = tid >> 5;
    const int rowIn = lane & 15;       // M row within tile
    const int khalf = lane >> 4;       // selects K pair {0,1} vs {2,3}
    const int mgrp  = blockIdx.x & (NUM_MGRP - 1);
    const int kg    = blockIdx.x >> 5; // K split index, grid = 32*10
    const int rowBase = mgrp * MGRP_ROWS;
    const int kstart  = kg * KRANGE;

    v8f acc[4] = {};

    // Stage chunk (global -> LDS) entirely via the Tensor Data Mover.
    auto stage = [&](int kbase, int buf) {
        // 64x32 fp32 tile of x, strided rows, HW-padded to 36-dword LDS rows
        tdm_load((unsigned)(uintptr_t)&ldsA[buf][0],
                 x + (size_t)rowBase * K_DIM + kbase,
                 /*tile_d0=*/KC, /*tile_d1=*/MGRP_ROWS,
                 /*tensor_d0=*/K_DIM, /*tensor_d1=*/B_ROWS,
                 /*stride0=*/(unsigned long long)K_DIM,
                 G1_XTILE);
        // contiguous 2048-float WMMA B-fragment chunk
        tdm_load((unsigned)(uintptr_t)&ldsB[buf][0],
                 kfrag + (size_t)kbase * NPAD,
                 /*tile_d0=*/2048, /*tile_d1=*/0,
                 /*tensor_d0=*/2048, /*tensor_d1=*/0,
                 /*stride0=*/2048ull,
                 G1_LINEAR);
    };

    auto compute = [&](int buf) {
#pragma unroll
        for (int kk = 0; kk < KC / 4; ++kk) {
            v2f a = *(const v2f*)&ldsA[buf][(wave * 16 + rowIn) * LDS_STRIDE + kk * 4 + khalf * 2];
#pragma unroll
            for (int nt = 0; nt < 4; ++nt) {
                v2f b = *(const v2f*)&ldsB[buf][((kk * 4 + nt) * 32 + lane) * 2];
                acc[nt] = __builtin_amdgcn_wmma_f32_16x16x4_f32(
                    false, a, false, b, (short)0, acc[nt], false, false);
            }
        }
    };

    int buf = 0;
    if (wave == 0) {
        stage(kstart, 0);
        __builtin_amdgcn_s_wait_tensorcnt(0);
    }
    __syncthreads();
    for (int c = 0; c < NCHUNK; ++c) {
        if (wave == 0 && (c + 1 < NCHUNK)) stage(kstart + (c + 1) * KC, buf ^ 1);
        compute(buf);
        if (wave == 0) __builtin_amdgcn_s_wait_tensorcnt(0);
        __syncthreads();
        buf ^= 1;
    }

    // C/D fragment layout: VGPR v -> M = v (lanes 0-15) / v+8 (lanes 16-31), N = lane&15
    float* base = pPart + (size_t)kg * B_ROWS * NPAD;
#pragma unroll
    for (int nt = 0; nt < 4; ++nt) {
#pragma unroll
        for (int v = 0; v < 8; ++v) {
            int mrow = rowBase + wave * 16 + v + khalf * 8;
            base[(size_t)mrow * NPAD + nt * 16 + rowIn] = acc[nt][v];
        }
    }
}

// ---------------------------------------------------------------------------
// K3: reduce the KSPLIT partials
// ---------------------------------------------------------------------------
__global__ void k_reduce(float* __restrict__ ws) {
    const float* pPart = ws + OFF_PPART;
    float* p           = ws + OFF_P;
    int i = blockIdx.x * 256 + threadIdx.x;  // < 131072
    float s = 0.f;
#pragma unroll
    for (int kg = 0; kg < KSPLIT; ++kg)
        s += pPart[(size_t)kg * B_ROWS * NPAD + i];
    p[i] = s;
}

// ---------------------------------------------------------------------------
// K4: closed-form inter-batch scan: last[b] = p[b] + d^100 * last[b-1]
// ---------------------------------------------------------------------------
__global__ void k_scan(float* __restrict__ ws) {
    const float* dpow = ws + OFF_DPOW;
    const float* p    = ws + OFF_P;
    float* last       = ws + OFF_LAST;
    int r = threadIdx.x;               // 0..63
    float q = (r < RES) ? dpow[r * 101 + 100] : 0.f;
    float s = 0.f;
    for (int b0 = 0; b0 < B_ROWS; b0 += 8) {
        float pv[8];
#pragma unroll
        for (int i = 0; i < 8; ++i) pv[i] = p[(size_t)(b0 + i) * NPAD + r];
#pragma unroll
        for (int i = 0; i < 8; ++i) {
            s = fmaf(q, s, pv[i]);
            last[(size_t)(b0 + i) * NPAD + r] = s;
        }
    }
}

// ---------------------------------------------------------------------------
// K5: out = last @ W_out^T   (2048x96, K=50 — tiny, VALU)
// ---------------------------------------------------------------------------
__global__ void k_out(const float* __restrict__ ws_last,
                      const float* __restrict__ W_out,
                      float* __restrict__ out) {
    int i = blockIdx.x * 256 + threadIdx.x;   // < 196608
    int b = i / PRED, pc = i % PRED;
    const float* lrow = ws_last + (size_t)b * NPAD;
    float s = 0.f;
#pragma unroll 10
    for (int r = 0; r < RES; ++r)
        s = fmaf(lrow[r], W_out[pc * RES + r], s);
    out[i] = s;
}

// ---------------------------------------------------------------------------
extern "C" void kernel_launch(void* const* d_in, const int* in_sizes, int n_in,
                              void* d_out, int out_size, void* d_ws, size_t ws_size,
                              hipStream_t stream) {
    const float* x     = (const float*)d_in[0];
    const float* W_lin = (const float*)d_in[1];
    const float* W_in  = (const float*)d_in[2];
    const float* d     = (const float*)d_in[3];
    const float* W_out = (const float*)d_in[4];
    float* out = (float*)d_out;
    float* ws  = (float*)d_ws;

    k_prep <<<1, 256, 0, stream>>>(W_lin, W_in, d, ws);
    k_kfrag<<<300, 256, 0, stream>>>(ws);
    k_gemm <<<NUM_MGRP * KSPLIT, 128, 0, stream>>>(x, ws + OFF_KFRAG, ws + OFF_PPART);
    k_reduce<<<512, 256, 0, stream>>>(ws);
    k_scan <<<1, 64, 0, stream>>>(ws);
    k_out  <<<768, 256, 0, stream>>>(ws + OFF_LAST, W_out, out);
}